// SimpleGCN_10453950399195
// MI455X (gfx1250) — compile-verified
//
#include <hip/hip_runtime.h>

#define N_NODES 50000
#define N_EDGES 600000
#define IN_CH   128
#define HID1    256
#define HID2    128
#define OUT_CH  2

typedef __attribute__((ext_vector_type(2))) float v2f;
typedef __attribute__((ext_vector_type(8))) float v8f;

// ---------------- degree / normalization ----------------

__global__ void k_deg_init(float* __restrict__ deg) {
    int n = blockIdx.x * blockDim.x + threadIdx.x;
    if (n < N_NODES) deg[n] = 1.0f;   // self-loop contributes 1
}

__global__ void k_deg_edges(const long long* __restrict__ ei, float* __restrict__ deg) {
    int e = blockIdx.x * blockDim.x + threadIdx.x;
    if (e < N_EDGES) {
        long long d = ei[N_EDGES + e];   // dst row of edge_index
        atomicAdd(&deg[d], 1.0f);
    }
}

__global__ void k_dinv(float* __restrict__ deg) {
    int n = blockIdx.x * blockDim.x + threadIdx.x;
    if (n < N_NODES) deg[n] = rsqrtf(deg[n]);   // deg >= 1 always
}

// ---------------- fp32 WMMA GEMM: out[N x M] = H[N x K] @ W[K x M] ----------------
// One wave per 16x64 output strip = 4 WMMA tiles along M.  The A fragment is
// loaded once per K-step (one b64 per lane) and reused by 4 v_wmma issues,
// cutting A-side L2 traffic 4x vs one-tile-per-wave.  K stepped by 4 via
// V_WMMA_F32_16X16X4_F32.
// A-frag (16x4): lanes 0-15 hold K=0,1; lanes 16-31 hold K=2,3 (row = lane&15).
// B-frag (4x16): transposed analogue.  C/D: 8 VGPRs, lanes 0-15 rows 0-7,
// lanes 16-31 rows 8-15, col = lane&15.

__global__ void __launch_bounds__(32)
k_gemm_wmma(const float* __restrict__ H, const float* __restrict__ W,
            float* __restrict__ out, int K, int M) {
    const int lane = threadIdx.x;
    const int half = lane >> 4;      // 0 or 1
    const int l    = lane & 15;
    const long long row  = (long long)blockIdx.x * 16 + l;   // node (A row)
    const int       col0 = blockIdx.y * 64 + l;              // first strip column

    v8f acc[4] = {};
    for (int k0 = 0; k0 < K; k0 += 4) {
        const int ka = k0 + half * 2;
        v2f a;
        a.x = H[row * K + ka];
        a.y = H[row * K + ka + 1];
        const float* w0 = W + (long long)ka * M + col0;
        const float* w1 = w0 + M;
#pragma unroll
        for (int t = 0; t < 4; ++t) {
            v2f b;
            b.x = w0[t * 16];
            b.y = w1[t * 16];
            acc[t] = __builtin_amdgcn_wmma_f32_16x16x4_f32(
                /*neg_a=*/false, a, /*neg_b=*/false, b,
                /*c_mod=*/(short)0, acc[t], /*reuse_a=*/false, /*reuse_b=*/false);
        }
    }

    const long long mbase = (long long)blockIdx.x * 16 + half * 8;
#pragma unroll
    for (int t = 0; t < 4; ++t) {
        float* op = out + mbase * M + col0 + t * 16;
#pragma unroll
        for (int r = 0; r < 8; ++r)
            op[(long long)r * M] = acc[t][r];
    }
}

// ---------------- tiny GEMM for layer 3 (M = 2) ----------------

__global__ void k_gemm_small(const float* __restrict__ H, const float* __restrict__ W3,
                             float* __restrict__ out) {
    int n = blockIdx.x * blockDim.x + threadIdx.x;
    if (n >= N_NODES) return;
    float s0 = 0.f, s1 = 0.f;
    const float* hp = H + (long long)n * HID2;
#pragma unroll 4
    for (int k = 0; k < HID2; ++k) {
        float h = hp[k];
        s0 += h * W3[k * 2 + 0];
        s1 += h * W3[k * 2 + 1];
    }
    out[n * 2 + 0] = s0;
    out[n * 2 + 1] = s1;
}

// ---------------- edge gather + scale + scatter (segment_sum via atomics) ----------------
// Work item = (edge-or-selfloop, VEC-channel chunk).  e >= N_EDGES encodes the
// self-loop of node (e - N_EDGES).  norm = dinv[src]*dinv[dst] on the fly.
// Gather is a single b128/b64 load; scatter is fp32 L2 atomics (L2-resident:
// every h buffer fits in the 192 MB L2).

template <int VEC>
__global__ void k_scatter(const float* __restrict__ h, const float* __restrict__ dinv,
                          const long long* __restrict__ ei, float* __restrict__ agg,
                          int M) {
    const int chunks = M / VEC;
    long long tid = (long long)blockIdx.x * blockDim.x + threadIdx.x;
    long long total = (long long)(N_EDGES + N_NODES) * chunks;
    if (tid >= total) return;
    int e = (int)(tid / chunks);
    int c = (int)(tid % chunks) * VEC;
    long long s, d;
    if (e < N_EDGES) { s = ei[e]; d = ei[N_EDGES + e]; }
    else             { s = d = e - N_EDGES; }
    const float w = dinv[s] * dinv[d];
    const float* hp = h + s * (long long)M + c;
    float* ap = agg + d * (long long)M + c;

    float v[VEC];
    if (VEC == 4) {
        float4 t = *reinterpret_cast<const float4*>(hp);
        v[0] = t.x; v[1] = t.y; v[2] = t.z; v[3] = t.w;
    } else {
        float2 t = *reinterpret_cast<const float2*>(hp);
        v[0] = t.x; v[1] = t.y;
    }
#pragma unroll
    for (int i = 0; i < VEC; ++i)
        atomicAdd(ap + i, v[i] * w);
}

// ---------------- bias + relu ----------------

template <int M>
__global__ void k_bias_relu(float* __restrict__ h, const float* __restrict__ b,
                            long long total) {
    long long t = (long long)blockIdx.x * blockDim.x + threadIdx.x;
    if (t >= total) return;
    float v = h[t] + b[(int)(t & (M - 1))];
    h[t] = v > 0.f ? v : 0.f;
}

// ---------------- bias + log_softmax over 2 classes ----------------

__global__ void k_bias_logsoftmax(const float* __restrict__ agg, const float* __restrict__ b3,
                                  float* __restrict__ out) {
    int n = blockIdx.x * blockDim.x + threadIdx.x;
    if (n >= N_NODES) return;
    float z0 = agg[n * 2 + 0] + b3[0];
    float z1 = agg[n * 2 + 1] + b3[1];
    float m  = fmaxf(z0, z1);
    float lse = m + logf(expf(z0 - m) + expf(z1 - m));
    out[n * 2 + 0] = z0 - lse;
    out[n * 2 + 1] = z1 - lse;
}

// ---------------- launch ----------------

extern "C" void kernel_launch(void* const* d_in, const int* in_sizes, int n_in,
                              void* d_out, int out_size, void* d_ws, size_t ws_size,
                              hipStream_t stream) {
    const float*     x  = (const float*)d_in[0];
    const float*     W1 = (const float*)d_in[1];
    const float*     b1 = (const float*)d_in[2];
    const float*     W2 = (const float*)d_in[3];
    const float*     b2 = (const float*)d_in[4];
    const float*     W3 = (const float*)d_in[5];
    const float*     b3 = (const float*)d_in[6];
    const long long* ei = (const long long*)d_in[7];   // int64 [2, E] row-major
    float* out = (float*)d_out;

    // workspace layout (fp32): buf0[N*256] | buf1[N*256] | dinv[N] | agg3[N*2]
    float* buf0 = (float*)d_ws;
    float* buf1 = buf0 + (size_t)N_NODES * HID1;
    float* dinv = buf1 + (size_t)N_NODES * HID1;
    float* agg3 = dinv + N_NODES;

    const int B = 256;
    const int gN = (N_NODES + B - 1) / B;
    const int gE = (N_EDGES + B - 1) / B;

    // symmetric normalization dinv = rsqrt(in-degree + 1)
    k_deg_init<<<gN, B, 0, stream>>>(dinv);
    k_deg_edges<<<gE, B, 0, stream>>>(ei, dinv);
    k_dinv<<<gN, B, 0, stream>>>(dinv);

    // ---- layer 1: x[N,128] @ W1[128,256] -> scatter -> +b1, relu ----
    {
        dim3 grid(N_NODES / 16, HID1 / 64);
        k_gemm_wmma<<<grid, 32, 0, stream>>>(x, W1, buf0, IN_CH, HID1);
        hipMemsetAsync(buf1, 0, sizeof(float) * (size_t)N_NODES * HID1, stream);
        long long tw = (long long)(N_EDGES + N_NODES) * (HID1 / 4);
        k_scatter<4><<<(int)((tw + B - 1) / B), B, 0, stream>>>(buf0, dinv, ei, buf1, HID1);
        long long tn = (long long)N_NODES * HID1;
        k_bias_relu<HID1><<<(int)((tn + B - 1) / B), B, 0, stream>>>(buf1, b1, tn);
    }

    // ---- layer 2: h1[N,256] @ W2[256,128] -> scatter -> +b2, relu ----
    {
        dim3 grid(N_NODES / 16, HID2 / 64);
        k_gemm_wmma<<<grid, 32, 0, stream>>>(buf1, W2, buf0, HID1, HID2);
        hipMemsetAsync(buf1, 0, sizeof(float) * (size_t)N_NODES * HID2, stream);
        long long tw = (long long)(N_EDGES + N_NODES) * (HID2 / 4);
        k_scatter<4><<<(int)((tw + B - 1) / B), B, 0, stream>>>(buf0, dinv, ei, buf1, HID2);
        long long tn = (long long)N_NODES * HID2;
        k_bias_relu<HID2><<<(int)((tn + B - 1) / B), B, 0, stream>>>(buf1, b2, tn);
    }

    // ---- layer 3: h2[N,128] @ W3[128,2] -> scatter -> +b3, log_softmax ----
    {
        k_gemm_small<<<gN, B, 0, stream>>>(buf1, W3, buf0);
        hipMemsetAsync(agg3, 0, sizeof(float) * (size_t)N_NODES * OUT_CH, stream);
        long long tw = (long long)(N_EDGES + N_NODES);   // VEC=2 covers both channels
        k_scatter<2><<<(int)((tw + B - 1) / B), B, 0, stream>>>(buf0, dinv, ei, agg3, OUT_CH);
        k_bias_logsoftmax<<<gN, B, 0, stream>>>(agg3, b3, out);
    }
}